// GraphAttentionRiskModel_19585050870159
// MI455X (gfx1250) — compile-verified
//
#include <hip/hip_runtime.h>
#include <hip/hip_bf16.h>

// ---------------- problem constants (from the reference) ----------------
#define NN      50000
#define EE      1600000
#define ETOT    (EE + NN)        // edges + self loops
#define IN_DIM  64
#define HID     16
#define HEADS   4
#define OUT_DIM 16
#define NEG_SLOPE 0.2f
#define EPSV    1e-16f

typedef __attribute__((ext_vector_type(2))) float v2f;
typedef __attribute__((ext_vector_type(8))) float v8f;

// ---------------- helpers ----------------
__device__ __forceinline__ v8f wmma_f32_k4(v2f a, v2f b, v8f c) {
  // V_WMMA_F32_16X16X4_F32 : D = A(16x4) x B(4x16) + C(16x16), all f32
  return __builtin_amdgcn_wmma_f32_16x16x4_f32(
      /*neg_a=*/false, a, /*neg_b=*/false, b,
      /*c_mod=*/(short)0, c, /*reuse_a=*/false, /*reuse_b=*/false);
}

__device__ __forceinline__ float lrelu(float v) {
  return v > 0.0f ? v : NEG_SLOPE * v;
}

__device__ __forceinline__ void atomicMaxF(float* addr, float val) {
  int* ai = (int*)addr;
  int old = __float_as_int(*addr);
  while (__int_as_float(old) < val) {
    int assumed = old;
    old = atomicCAS(ai, assumed, __float_as_int(val));
    if (old == assumed) break;
  }
}

__device__ __forceinline__ void edge_sd(const long long* __restrict__ ei,
                                        int e, int* s, int* d) {
  if (e < EE) {
    *s = (int)ei[e];
    *d = (int)ei[(long long)EE + e];
  } else {            // synthesized self loop
    *s = e - EE;
    *d = e - EE;
  }
}

// ---------------- fill ----------------
__global__ void fill_kernel(float* __restrict__ p, int n, float v) {
  int i = blockIdx.x * blockDim.x + threadIdx.x;
  if (i < n) p[i] = v;
}

// ---------------- GEMM 1: h1[N,64] = x[N,64] @ W1[64,64] (WMMA f32) ------
// grid = N/16 row tiles, 4 waves per block (one 16-col tile per wave)
__global__ __launch_bounds__(128) void gemm1_wmma(const float* __restrict__ x,
                                                  const float* __restrict__ W,
                                                  float* __restrict__ h) {
  const int lane = threadIdx.x & 31;
  const int wave = threadIdx.x >> 5;          // col tile 0..3
  const int r    = lane & 15;
  const int half = lane >> 4;                 // 0 or 1
  const int row0 = blockIdx.x * 16;
  const int c0   = wave * 16;
  const float* xr = x + (row0 + r) * IN_DIM;
  v8f acc = {};
#pragma unroll
  for (int kk = 0; kk < 16; ++kk) {
    const int kb = kk * 4 + half * 2;
    v2f a, b;
    a.x = xr[kb + 0];
    a.y = xr[kb + 1];
    b.x = W[(kb + 0) * 64 + c0 + r];
    b.y = W[(kb + 1) * 64 + c0 + r];
    acc = wmma_f32_k4(a, b, acc);
  }
#pragma unroll
  for (int i = 0; i < 8; ++i)
    h[(row0 + half * 8 + i) * 64 + c0 + r] = acc[i];
}

// ---------------- GEMM 2: h2[N,16] = hact[N,64] @ W2[64,16] (WMMA f32) ---
__global__ __launch_bounds__(32) void gemm2_wmma(const float* __restrict__ x,
                                                 const float* __restrict__ W,
                                                 float* __restrict__ h) {
  const int lane = threadIdx.x & 31;
  const int r    = lane & 15;
  const int half = lane >> 4;
  const int row0 = blockIdx.x * 16;
  const float* xr = x + (row0 + r) * 64;
  v8f acc = {};
#pragma unroll
  for (int kk = 0; kk < 16; ++kk) {
    const int kb = kk * 4 + half * 2;
    v2f a, b;
    a.x = xr[kb + 0];
    a.y = xr[kb + 1];
    b.x = W[(kb + 0) * 16 + r];
    b.y = W[(kb + 1) * 16 + r];
    acc = wmma_f32_k4(a, b, acc);
  }
#pragma unroll
  for (int i = 0; i < 8; ++i)
    h[(row0 + half * 8 + i) * 16 + r] = acc[i];
}

// ----- per-node attention logits: al[n,h] = dot(h[n,h,:], a[h,:]) --------
__global__ void attn_logits_kernel(const float* __restrict__ h,
                                   const float* __restrict__ a_src,
                                   const float* __restrict__ a_dst,
                                   float* __restrict__ als,
                                   float* __restrict__ ald,
                                   int heads, int ch) {
  int i = blockIdx.x * blockDim.x + threadIdx.x;
  if (i >= NN * heads) return;
  const int node = i / heads, hh = i % heads;
  const float* hp = h + (node * heads + hh) * ch;
  const float* as = a_src + hh * ch;
  const float* ad = a_dst + hh * ch;
  float s0 = 0.0f, s1 = 0.0f;
  for (int c = 0; c < ch; ++c) {
    float v = hp[c];
    s0 += v * as[c];
    s1 += v * ad[c];
  }
  als[i] = s0;
  ald[i] = s1;
}

// ----- edge pass 1: segment max -----------------------------------------
__global__ void edge_max_kernel(const long long* __restrict__ ei,
                                const float* __restrict__ als,
                                const float* __restrict__ ald,
                                float* __restrict__ m, int heads) {
  long long t = (long long)blockIdx.x * blockDim.x + threadIdx.x;
  if (t >= (long long)ETOT * heads) return;
  int e  = (int)(t / heads);
  int hh = (int)(t % heads);
  int s, d;
  edge_sd(ei, e, &s, &d);
  float ev = lrelu(als[s * heads + hh] + ald[d * heads + hh]);
  atomicMaxF(&m[d * heads + hh], ev);
}

// ----- edge pass 2: segment sum of exp(e - m) ----------------------------
__global__ void edge_sum_kernel(const long long* __restrict__ ei,
                                const float* __restrict__ als,
                                const float* __restrict__ ald,
                                const float* __restrict__ m,
                                float* __restrict__ ssum, int heads) {
  long long t = (long long)blockIdx.x * blockDim.x + threadIdx.x;
  if (t >= (long long)ETOT * heads) return;
  int e  = (int)(t / heads);
  int hh = (int)(t % heads);
  int s, d;
  edge_sd(ei, e, &s, &d);
  float ev = lrelu(als[s * heads + hh] + ald[d * heads + hh]);
  float ex = expf(ev - m[d * heads + hh]);
  atomicAdd(&ssum[d * heads + hh], ex);
}

// ----- edge pass 3: out[dst] += alpha * h[src] ---------------------------
__global__ void edge_agg_kernel(const long long* __restrict__ ei,
                                const float* __restrict__ als,
                                const float* __restrict__ ald,
                                const float* __restrict__ m,
                                const float* __restrict__ ssum,
                                const float* __restrict__ h,
                                float* __restrict__ out,
                                int heads, int ch) {
  int e = blockIdx.x * blockDim.x + threadIdx.x;
  if (e >= ETOT) return;
  int s, d;
  edge_sd(ei, e, &s, &d);
  for (int hh = 0; hh < heads; ++hh) {
    float ev    = lrelu(als[s * heads + hh] + ald[d * heads + hh]);
    float ex    = expf(ev - m[d * heads + hh]);
    float alpha = ex / (ssum[d * heads + hh] + EPSV);
    const float* hp = h + (s * heads + hh) * ch;
    float* op       = out + (d * heads + hh) * ch;
    for (int c = 0; c < ch; ++c)
      atomicAdd(&op[c], hp[c] * alpha);
  }
}

// ----- bias + ELU (between layers) ---------------------------------------
__global__ void bias_elu_kernel(const float* __restrict__ acc,
                                const float* __restrict__ b,
                                float* __restrict__ out, int feat) {
  int i = blockIdx.x * blockDim.x + threadIdx.x;
  if (i >= NN * feat) return;
  float v = acc[i] + b[i % feat];
  out[i] = v > 0.0f ? v : (expf(v) - 1.0f);
}

// ----- bias + sigmoid (final) --------------------------------------------
__global__ void bias_sigmoid_kernel(const float* __restrict__ acc,
                                    const float* __restrict__ b,
                                    float* __restrict__ out, int feat) {
  int i = blockIdx.x * blockDim.x + threadIdx.x;
  if (i >= NN * feat) return;
  float v = acc[i] + b[i % feat];
  out[i] = 1.0f / (1.0f + expf(-v));
}

// =========================================================================
extern "C" void kernel_launch(void* const* d_in, const int* in_sizes, int n_in,
                              void* d_out, int out_size, void* d_ws, size_t ws_size,
                              hipStream_t stream) {
  const float*     x      = (const float*)d_in[0];
  const long long* ei     = (const long long*)d_in[1];   // int64 [2,E]
  const float*     W1     = (const float*)d_in[2];
  const float*     a_src1 = (const float*)d_in[3];
  const float*     a_dst1 = (const float*)d_in[4];
  const float*     b1     = (const float*)d_in[5];
  const float*     W2     = (const float*)d_in[6];
  const float*     a_src2 = (const float*)d_in[7];
  const float*     a_dst2 = (const float*)d_in[8];
  const float*     b2     = (const float*)d_in[9];
  float*           out    = (float*)d_out;

  // workspace carve-up (floats)
  float* ws   = (float*)d_ws;
  float* h1   = ws;            ws += NN * 64;   // layer1 projection
  float* alS1 = ws;            ws += NN * HEADS;
  float* alD1 = ws;            ws += NN * HEADS;
  float* m1   = ws;            ws += NN * HEADS;
  float* s1   = ws;            ws += NN * HEADS;
  float* hact = ws;            ws += NN * 64;   // layer1 out -> ELU -> layer2 in
  float* h2   = ws;            ws += NN * OUT_DIM;
  float* alS2 = ws;            ws += NN;
  float* alD2 = ws;            ws += NN;
  float* m2   = ws;            ws += NN;
  float* s2   = ws;            ws += NN;
  float* agg2 = ws;            ws += NN * OUT_DIM;

  const int TB = 256;
  const float NEG_INF = -__builtin_huge_valf();

  // ---- init accumulators (deterministic each call) ----
  fill_kernel<<<(NN * 64 + TB - 1) / TB, TB, 0, stream>>>(hact, NN * 64, 0.0f);
  fill_kernel<<<(NN * OUT_DIM + TB - 1) / TB, TB, 0, stream>>>(agg2, NN * OUT_DIM, 0.0f);
  fill_kernel<<<(NN * HEADS + TB - 1) / TB, TB, 0, stream>>>(s1, NN * HEADS, 0.0f);
  fill_kernel<<<(NN + TB - 1) / TB, TB, 0, stream>>>(s2, NN, 0.0f);
  fill_kernel<<<(NN * HEADS + TB - 1) / TB, TB, 0, stream>>>(m1, NN * HEADS, NEG_INF);
  fill_kernel<<<(NN + TB - 1) / TB, TB, 0, stream>>>(m2, NN, NEG_INF);

  // ---- layer 1 ----
  gemm1_wmma<<<NN / 16, 128, 0, stream>>>(x, W1, h1);            // 3125 blocks
  attn_logits_kernel<<<(NN * HEADS + TB - 1) / TB, TB, 0, stream>>>(
      h1, a_src1, a_dst1, alS1, alD1, HEADS, HID);

  {
    long long t1 = (long long)ETOT * HEADS;
    int g1 = (int)((t1 + TB - 1) / TB);
    int ge = (ETOT + TB - 1) / TB;
    edge_max_kernel<<<g1, TB, 0, stream>>>(ei, alS1, alD1, m1, HEADS);
    edge_sum_kernel<<<g1, TB, 0, stream>>>(ei, alS1, alD1, m1, s1, HEADS);
    edge_agg_kernel<<<ge, TB, 0, stream>>>(ei, alS1, alD1, m1, s1, h1, hact, HEADS, HID);
  }
  bias_elu_kernel<<<(NN * 64 + TB - 1) / TB, TB, 0, stream>>>(hact, b1, hact, 64);

  // ---- layer 2 ----
  gemm2_wmma<<<NN / 16, 32, 0, stream>>>(hact, W2, h2);
  attn_logits_kernel<<<(NN + TB - 1) / TB, TB, 0, stream>>>(
      h2, a_src2, a_dst2, alS2, alD2, 1, OUT_DIM);

  {
    int g1 = (ETOT + TB - 1) / TB;
    edge_max_kernel<<<g1, TB, 0, stream>>>(ei, alS2, alD2, m2, 1);
    edge_sum_kernel<<<g1, TB, 0, stream>>>(ei, alS2, alD2, m2, s2, 1);
    edge_agg_kernel<<<g1, TB, 0, stream>>>(ei, alS2, alD2, m2, s2, h2, agg2, 1, OUT_DIM);
  }
  bias_sigmoid_kernel<<<(NN * OUT_DIM + TB - 1) / TB, TB, 0, stream>>>(agg2, b2, out, OUT_DIM);
}